// HandcraftFeature_lamb_31808527794660
// MI455X (gfx1250) — compile-verified
//
#include <hip/hip_runtime.h>
#include <hip/hip_bf16.h>

typedef __attribute__((ext_vector_type(2))) float v2f;
typedef __attribute__((ext_vector_type(8))) float v8f;

#define GF_EPS 1e-4f

// D = A(16x4,f32) * B(4x16,f32) + C(16x16,f32), exact f32 via matrix pipe.
__device__ __forceinline__ v8f wmma4(v2f a, v2f b, v8f c) {
    return __builtin_amdgcn_wmma_f32_16x16x4_f32(false, a, false, b, (short)0, c,
                                                 false, false);
}

__device__ __forceinline__ int clampi(int v, int lo, int hi) {
    return v < lo ? lo : (v > hi ? hi : v);
}

// ---------------------------------------------------------------------------
// Stage 1: per-pixel costd + center plane.
// One block = 16x16 spatial tile of one batch. LDS holds 49 views x 18x18 halo.
// ---------------------------------------------------------------------------
__global__ __launch_bounds__(256)
void costd_kernel(const float* __restrict__ x, float* __restrict__ Xp,
                  float* __restrict__ Yp, int H, int W)
{
    __shared__ float sv[49][324]; // 18*18 per view, 63504 B
    const int tx = threadIdx.x, ty = threadIdx.y;
    const int tid = ty * 16 + tx;
    const int w0 = blockIdx.x * 16, h0 = blockIdx.y * 16, b = blockIdx.z;
    const size_t plane = (size_t)H * W;

    for (int idx = tid; idx < 49 * 324; idx += 256) {
        int view = idx / 324;
        int rem  = idx - view * 324;
        int ly = rem / 18;
        int lx = rem - ly * 18;
        int gh = h0 - 1 + ly;
        int gw = w0 - 1 + lx;
        float val = 0.f;
        if ((unsigned)gh < (unsigned)H && (unsigned)gw < (unsigned)W)
            val = x[(size_t)(b * 49 + view) * plane + (size_t)gh * W + gw];
        sv[view][rem] = val;
    }
    __syncthreads();

    const int cc = (ty + 1) * 18 + (tx + 1);
    float num = 0.f, den = 0.f;
    for (int u = 0; u < 7; ++u) {
        for (int v = 0; v < 7; ++v) {
            // angular neighbors with zero pad (lenslet Sobel)
            float am1m1 = (u > 0 && v > 0) ? sv[(u - 1) * 7 + (v - 1)][cc] : 0.f;
            float am10  = (u > 0)          ? sv[(u - 1) * 7 + v][cc]       : 0.f;
            float am1p1 = (u > 0 && v < 6) ? sv[(u - 1) * 7 + (v + 1)][cc] : 0.f;
            float a0m1  = (v > 0)          ? sv[u * 7 + (v - 1)][cc]       : 0.f;
            float a0p1  = (v < 6)          ? sv[u * 7 + (v + 1)][cc]       : 0.f;
            float ap1m1 = (u < 6 && v > 0) ? sv[(u + 1) * 7 + (v - 1)][cc] : 0.f;
            float ap10  = (u < 6)          ? sv[(u + 1) * 7 + v][cc]       : 0.f;
            float ap1p1 = (u < 6 && v < 6) ? sv[(u + 1) * 7 + (v + 1)][cc] : 0.f;
            // gx: gradient along v (columns); gy: along u (rows)
            float glx = (am1m1 + 2.f * a0m1 + ap1m1) - (am1p1 + 2.f * a0p1 + ap1p1);
            float gly = (am1m1 + 2.f * am10 + am1p1) - (ap1m1 + 2.f * ap10 + ap1p1);
            // spatial Sobel of this view (halo already zero-padded)
            const float* t = sv[u * 7 + v];
            float qmm = t[cc - 19], qm0 = t[cc - 18], qmp = t[cc - 17];
            float q0m = t[cc - 1],                    q0p = t[cc + 1];
            float qpm = t[cc + 17], qp0 = t[cc + 18], qpp = t[cc + 19];
            float gsx = (qmm + 2.f * q0m + qpm) - (qmp + 2.f * q0p + qpp);
            float gsy = (qmm + 2.f * qm0 + qmp) - (qpm + 2.f * qp0 + qpp);
            float p0 = glx * gsx, p1 = gly * gsy;
            float s0 = (p0 > 0.f) ? 1.f : ((p0 < 0.f) ? -1.f : 0.f);
            float s1 = (p1 > 0.f) ? 1.f : ((p1 < 0.f) ? -1.f : 0.f);
            float lam = (s0 == s1) ? 1.f : 0.f;
            num += lam * s0;
            den += lam;
        }
    }
    float nc = num / 49.f, dc = den / 49.f;
    float cost = (dc == 0.f) ? 0.f : nc / dc;
    size_t o = ((size_t)b * H + (h0 + ty)) * W + (w0 + tx);
    Xp[o] = sv[3 * 7 + 3][cc]; // x_center
    Yp[o] = cost;
}

// ---------------------------------------------------------------------------
// Stage 2: guided filter. 9-tap box sums as banded-matrix WMMA (exact f32).
// Horizontal: out(16x16) = In(16x24) x Band(24x16); six K=4 WMMA steps.
// A layout: lane<16 -> K {0,1}, lane>=16 -> K {2,3}; B mirrored; D: vgpr i ->
// rows i (lanes 0-15) / i+8 (lanes 16-31). One wave per tile, EXEC all-1.
// Edge handling is branchless: clamped address + 0/1 mask multiply.
// ---------------------------------------------------------------------------
__global__ __launch_bounds__(32)
void boxh4_kernel(const float* __restrict__ X, const float* __restrict__ Y,
                  float* __restrict__ Hx, float* __restrict__ Hy,
                  float* __restrict__ Hxy, float* __restrict__ Hxx, int W)
{
    const int tilesW = W >> 4;
    const int r0 = (blockIdx.x / tilesW) << 4;
    const int c0 = (blockIdx.x % tilesW) << 4;
    const int lane = threadIdx.x, half = lane >> 4, l15 = lane & 15;
    v8f aX = {}, aY = {}, aXY = {}, aXX = {};
    const size_t base = (size_t)(r0 + l15) * W;
    for (int k = 0; k < 6; ++k) {
        int kg = 4 * k + 2 * half;       // this lane's two K indices: kg, kg+1
        int cA = c0 - 4 + kg;            // input column of K index kg
        int ci0 = clampi(cA, 0, W - 1);
        int ci1 = clampi(cA + 1, 0, W - 1);
        float m0 = ((unsigned)cA < (unsigned)W) ? 1.f : 0.f;
        float m1 = ((unsigned)(cA + 1) < (unsigned)W) ? 1.f : 0.f;
        float x0 = X[base + ci0] * m0;
        float x1 = X[base + ci1] * m1;
        float y0 = Y[base + ci0] * m0;
        float y1 = Y[base + ci1] * m1;
        v2f Bm;                          // band: 1 iff n <= kg <= n+8
        Bm[0] = (kg >= l15 && kg <= l15 + 8) ? 1.f : 0.f;
        Bm[1] = (kg + 1 >= l15 && kg + 1 <= l15 + 8) ? 1.f : 0.f;
        v2f ax;  ax[0] = x0;       ax[1] = x1;
        v2f ay;  ay[0] = y0;       ay[1] = y1;
        v2f axy; axy[0] = x0 * y0; axy[1] = x1 * y1;
        v2f axx; axx[0] = x0 * x0; axx[1] = x1 * x1;
        aX  = wmma4(ax,  Bm, aX);
        aY  = wmma4(ay,  Bm, aY);
        aXY = wmma4(axy, Bm, aXY);
        aXX = wmma4(axx, Bm, aXX);
    }
    for (int i = 0; i < 8; ++i) {
        size_t o = (size_t)(r0 + i + 8 * half) * W + (c0 + l15);
        Hx[o] = aX[i]; Hy[o] = aY[i]; Hxy[o] = aXY[i]; Hxx[o] = aXX[i];
    }
}

__global__ __launch_bounds__(32)
void boxv_ab_kernel(const float* __restrict__ Hx, const float* __restrict__ Hy,
                    const float* __restrict__ Hxy, const float* __restrict__ Hxx,
                    float* __restrict__ Ap, float* __restrict__ Bp, int H, int W)
{
    const int tilesW = W >> 4;
    const int r0 = (blockIdx.x / tilesW) << 4;
    const int c0 = (blockIdx.x % tilesW) << 4;
    const int lane = threadIdx.x, half = lane >> 4, l15 = lane & 15;
    const int rlo = (r0 / H) * H, rhi = rlo + H;   // batch row range
    const int col = c0 + l15;
    v8f sX = {}, sY = {}, sXY = {}, sXX = {};
    for (int k = 0; k < 6; ++k) {
        int kg = 4 * k + 2 * half;
        v2f Am;                          // band: 1 iff m <= kg <= m+8
        Am[0] = (kg >= l15 && kg <= l15 + 8) ? 1.f : 0.f;
        Am[1] = (kg + 1 >= l15 && kg + 1 <= l15 + 8) ? 1.f : 0.f;
        int rr = r0 - 4 + kg;
        int r0c = clampi(rr, rlo, rhi - 1);
        int r1c = clampi(rr + 1, rlo, rhi - 1);
        float m0 = (rr >= rlo && rr < rhi) ? 1.f : 0.f;
        float m1 = (rr + 1 >= rlo && rr + 1 < rhi) ? 1.f : 0.f;
        size_t i0 = (size_t)r0c * W + col, i1 = (size_t)r1c * W + col;
        v2f bx;  bx[0]  = Hx[i0]  * m0; bx[1]  = Hx[i1]  * m1;
        v2f by;  by[0]  = Hy[i0]  * m0; by[1]  = Hy[i1]  * m1;
        v2f bxy; bxy[0] = Hxy[i0] * m0; bxy[1] = Hxy[i1] * m1;
        v2f bxx; bxx[0] = Hxx[i0] * m0; bxx[1] = Hxx[i1] * m1;
        sX  = wmma4(Am, bx,  sX);
        sY  = wmma4(Am, by,  sY);
        sXY = wmma4(Am, bxy, sXY);
        sXX = wmma4(Am, bxx, sXX);
    }
    float nw = (float)(min(col + 4, W - 1) - max(col - 4, 0) + 1);
    for (int i = 0; i < 8; ++i) {
        int r = r0 + i + 8 * half;
        int h = r - rlo;
        float nh = (float)(min(h + 4, H - 1) - max(h - 4, 0) + 1);
        float N = nh * nw;
        float mx = sX[i] / N, my = sY[i] / N;
        float cov = sXY[i] / N - mx * my;
        float var = sXX[i] / N - mx * mx;
        float Av = cov / (var + GF_EPS);
        float bv = my - Av * mx;
        size_t o = (size_t)r * W + col;
        Ap[o] = Av; Bp[o] = bv;
    }
}

__global__ __launch_bounds__(32)
void boxh2_kernel(const float* __restrict__ Ain, const float* __restrict__ Bin,
                  float* __restrict__ Ha, float* __restrict__ Hb, int W)
{
    const int tilesW = W >> 4;
    const int r0 = (blockIdx.x / tilesW) << 4;
    const int c0 = (blockIdx.x % tilesW) << 4;
    const int lane = threadIdx.x, half = lane >> 4, l15 = lane & 15;
    v8f aA = {}, aB = {};
    const size_t base = (size_t)(r0 + l15) * W;
    for (int k = 0; k < 6; ++k) {
        int kg = 4 * k + 2 * half;
        int cA = c0 - 4 + kg;
        int ci0 = clampi(cA, 0, W - 1);
        int ci1 = clampi(cA + 1, 0, W - 1);
        float m0 = ((unsigned)cA < (unsigned)W) ? 1.f : 0.f;
        float m1 = ((unsigned)(cA + 1) < (unsigned)W) ? 1.f : 0.f;
        v2f Bm;
        Bm[0] = (kg >= l15 && kg <= l15 + 8) ? 1.f : 0.f;
        Bm[1] = (kg + 1 >= l15 && kg + 1 <= l15 + 8) ? 1.f : 0.f;
        v2f va; va[0] = Ain[base + ci0] * m0; va[1] = Ain[base + ci1] * m1;
        v2f vb; vb[0] = Bin[base + ci0] * m0; vb[1] = Bin[base + ci1] * m1;
        aA = wmma4(va, Bm, aA);
        aB = wmma4(vb, Bm, aB);
    }
    for (int i = 0; i < 8; ++i) {
        size_t o = (size_t)(r0 + i + 8 * half) * W + (c0 + l15);
        Ha[o] = aA[i]; Hb[o] = aB[i];
    }
}

__global__ __launch_bounds__(32)
void boxv_out_kernel(const float* __restrict__ Ha, const float* __restrict__ Hb,
                     const float* __restrict__ Xp, float* __restrict__ out,
                     int H, int W)
{
    const int tilesW = W >> 4;
    const int r0 = (blockIdx.x / tilesW) << 4;
    const int c0 = (blockIdx.x % tilesW) << 4;
    const int lane = threadIdx.x, half = lane >> 4, l15 = lane & 15;
    const int rlo = (r0 / H) * H, rhi = rlo + H;
    const int col = c0 + l15;
    v8f sA = {}, sB = {};
    for (int k = 0; k < 6; ++k) {
        int kg = 4 * k + 2 * half;
        v2f Am;
        Am[0] = (kg >= l15 && kg <= l15 + 8) ? 1.f : 0.f;
        Am[1] = (kg + 1 >= l15 && kg + 1 <= l15 + 8) ? 1.f : 0.f;
        int rr = r0 - 4 + kg;
        int r0c = clampi(rr, rlo, rhi - 1);
        int r1c = clampi(rr + 1, rlo, rhi - 1);
        float m0 = (rr >= rlo && rr < rhi) ? 1.f : 0.f;
        float m1 = (rr + 1 >= rlo && rr + 1 < rhi) ? 1.f : 0.f;
        size_t i0 = (size_t)r0c * W + col, i1 = (size_t)r1c * W + col;
        v2f ba; ba[0] = Ha[i0] * m0; ba[1] = Ha[i1] * m1;
        v2f bb; bb[0] = Hb[i0] * m0; bb[1] = Hb[i1] * m1;
        sA = wmma4(Am, ba, sA);
        sB = wmma4(Am, bb, sB);
    }
    float nw = (float)(min(col + 4, W - 1) - max(col - 4, 0) + 1);
    for (int i = 0; i < 8; ++i) {
        int r = r0 + i + 8 * half;
        int h = r - rlo;
        float nh = (float)(min(h + 4, H - 1) - max(h - 4, 0) + 1);
        float N = nh * nw;
        size_t o = (size_t)r * W + col;
        out[o] = (sA[i] / N) * Xp[o] + (sB[i] / N);
    }
}

// ---------------------------------------------------------------------------
extern "C" void kernel_launch(void* const* d_in, const int* in_sizes, int n_in,
                              void* d_out, int out_size, void* d_ws, size_t ws_size,
                              hipStream_t stream) {
    (void)in_sizes; (void)n_in; (void)out_size; (void)ws_size;
    const float* x = (const float*)d_in[0];
    float* out = (float*)d_out;
    float* ws = (float*)d_ws;

    const int Bn = 2, H = 512, W = 512;
    const int R = Bn * H;                 // 1024 flattened rows
    const size_t P = (size_t)R * W;       // plane elems (= out_size)

    float* Xp  = ws + 0 * P;  // x_center
    float* Yp  = ws + 1 * P;  // costd
    float* Hx  = ws + 2 * P;
    float* Hy  = ws + 3 * P;
    float* Hxy = ws + 4 * P;
    float* Hxx = ws + 5 * P;
    float* Ap  = ws + 6 * P;
    float* Bp  = ws + 7 * P;
    float* Ha  = ws + 8 * P;
    float* Hb  = ws + 9 * P;

    dim3 blk1(16, 16);
    dim3 grd1(W / 16, H / 16, Bn);
    costd_kernel<<<grd1, blk1, 0, stream>>>(x, Xp, Yp, H, W);

    const int ntiles = (R / 16) * (W / 16); // 2048 tiles, one wave each
    boxh4_kernel  <<<ntiles, 32, 0, stream>>>(Xp, Yp, Hx, Hy, Hxy, Hxx, W);
    boxv_ab_kernel<<<ntiles, 32, 0, stream>>>(Hx, Hy, Hxy, Hxx, Ap, Bp, H, W);
    boxh2_kernel  <<<ntiles, 32, 0, stream>>>(Ap, Bp, Ha, Hb, W);
    boxv_out_kernel<<<ntiles, 32, 0, stream>>>(Ha, Hb, Xp, out, H, W);
}